// Conv2dCustom_44899588112769
// MI455X (gfx1250) — compile-verified
//
#include <hip/hip_runtime.h>

typedef __attribute__((ext_vector_type(2))) float v2f;
typedef __attribute__((ext_vector_type(8))) float v8f;

#define HH 512
#define WW 512
#define NIMG 16
#define CO 32
#define TH 16
#define TW 64
#define LDSP 68   // LDS row pitch (>= TW+2), padded

__global__ __launch_bounds__(256) void conv3x3_sum_wmma(
    const float* __restrict__ x, const float* __restrict__ wgt,
    const float* __restrict__ bias, float* __restrict__ out)
{
    __shared__ float xs[(TH + 2) * LDSP];

    const int n  = blockIdx.z;
    const int h0 = blockIdx.y * TH;
    const int w0 = blockIdx.x * TW;
    const int tid = threadIdx.x;

    const float* xn = x + (size_t)n * 3 * HH * WW;

    // ---- Stage channel-summed input tile (with 1-px zero halo) into LDS ----
    for (int i = tid; i < (TH + 2) * (TW + 2); i += 256) {
        const int r  = i / (TW + 2);
        const int c  = i - r * (TW + 2);
        const int gh = h0 - 1 + r;
        const int gw = w0 - 1 + c;
        float v = 0.0f;
        if ((unsigned)gh < (unsigned)HH && (unsigned)gw < (unsigned)WW) {
            const size_t off = (size_t)gh * WW + gw;
            v = xn[off] + xn[off + (size_t)HH * WW] + xn[off + 2 * (size_t)HH * WW];
        }
        xs[r * LDSP + c] = v;
    }
    __syncthreads();

    const int wave = tid >> 5;   // 8 waves per WG (wave32)
    const int lane = tid & 31;
    const int lh   = lane >> 4;  // lane half: +2 on K (A/B), +8 on M (C/D)
    const int ln   = lane & 15;  // M for A, N for B/C/D

    for (int cohalf = 0; cohalf < 2; ++cohalf) {
        // A fragments: a[c][j] = W[co = cohalf*16 + ln][k = 4c + 2*lh + j], 0 if k >= 9
        v2f a[3];
        #pragma unroll
        for (int c = 0; c < 3; ++c) {
            #pragma unroll
            for (int j = 0; j < 2; ++j) {
                const int k  = 4 * c + 2 * lh + j;
                const int kk = (k < 9) ? k : 0;                 // clamp: keep load in-bounds
                const float wv = wgt[(cohalf * 16 + ln) * 9 + kk];
                a[c][j] = (k < 9) ? wv : 0.0f;                  // cndmask, no divergence
            }
        }
        // Bias per output row: D VGPR v holds co = cohalf*16 + 8*lh + v
        float brow[8];
        #pragma unroll
        for (int v = 0; v < 8; ++v)
            brow[v] = bias[cohalf * 16 + 8 * lh + v];

        // 64 spatial tiles (16 rows x 4 sixteen-pixel segments), 8 per wave
        for (int t = wave; t < 64; t += 8) {
            const int hloc = t >> 2;
            const int wloc = (t & 3) * 16 + ln;

            v8f acc = {0.f, 0.f, 0.f, 0.f, 0.f, 0.f, 0.f, 0.f};
            #pragma unroll
            for (int c = 0; c < 3; ++c) {
                v2f b;
                #pragma unroll
                for (int j = 0; j < 2; ++j) {
                    const int k  = 4 * c + 2 * lh + j;
                    const int kk = (k < 9) ? k : 0;
                    const int kh = kk / 3;
                    const int kw = kk - kh * 3;
                    const float bv = xs[(hloc + kh) * LDSP + (wloc + kw)];
                    b[j] = (k < 9) ? bv : 0.0f;
                }
                // D = A x B + C   (16x16x4 fp32 WMMA)
                acc = __builtin_amdgcn_wmma_f32_16x16x4_f32(
                    false, a[c], false, b, (short)0, acc, false, false);
            }

            const int gh = h0 + hloc;
            const int gw = w0 + (t & 3) * 16 + ln;
            float* op = out + (((size_t)n * CO + cohalf * 16 + 8 * lh) * HH + gh) * WW + gw;
            #pragma unroll
            for (int v = 0; v < 8; ++v)
                op[(size_t)v * HH * WW] = acc[v] + brow[v];
        }
    }
}

extern "C" void kernel_launch(void* const* d_in, const int* in_sizes, int n_in,
                              void* d_out, int out_size, void* d_ws, size_t ws_size,
                              hipStream_t stream) {
    const float* x    = (const float*)d_in[0];   // (16,3,512,512)
    const float* wgt  = (const float*)d_in[1];   // (32,3,3)
    const float* bias = (const float*)d_in[2];   // (32,)
    float* out        = (float*)d_out;           // (16,32,512,512)

    dim3 grid(WW / TW, HH / TH, NIMG);           // (8, 32, 16)
    dim3 block(256);
    conv3x3_sum_wmma<<<grid, block, 0, stream>>>(x, wgt, bias, out);
}